// BiLSTM_CRF_5738076307914
// MI455X (gfx1250) — compile-verified
//
#include <hip/hip_runtime.h>

typedef __attribute__((ext_vector_type(2))) float v2f;
typedef __attribute__((ext_vector_type(8))) float v8f;

static constexpr int S = 512;
static constexpr int B = 256;
static constexpr int T = 128;

// ---------------------------------------------------------------------------
// Kernel 0: expTt[j*T + i] = exp(transitions[i*T + j])  (transposed exp table)
// ---------------------------------------------------------------------------
__global__ __launch_bounds__(256) void crf_exp_trans_kernel(
    const float* __restrict__ trans, float* __restrict__ expTt) {
  int idx = blockIdx.x * 256 + threadIdx.x;   // 0 .. T*T-1
  int i = idx >> 7;       // row of transitions
  int j = idx & 127;      // col of transitions
  expTt[j * T + i] = __expf(trans[idx]);
}

// ---------------------------------------------------------------------------
// Kernel 1: forward scan (normalizer). 16 blocks x 128 threads.
// Each block owns 16 batch rows; alpha kept in LDS across all 512 steps.
// B-fragments of exp(transitions) are hoisted into registers (loop-invariant
// across the whole scan), so the steady-state inner loop is ds_load + WMMA.
// ---------------------------------------------------------------------------
__global__ __launch_bounds__(128) void crf_forward_kernel(
    const float* __restrict__ em, const unsigned char* __restrict__ maskp,
    const float* __restrict__ startT, const float* __restrict__ endT,
    const float* __restrict__ expTt, float* __restrict__ normOut) {

  __shared__ __align__(16) float sAlpha[16][132];   // padded: stride 132 dwords
  __shared__ __align__(16) float sExpA[16][132];
  __shared__ __align__(16) float sEm[2][16][128];   // double-buffered emissions
  __shared__ float sPart[16][8];
  __shared__ float sAmax[16];
  __shared__ unsigned char sMask[16];

  const int tid  = threadIdx.x;
  const int lane = tid & 31;
  const int wave = tid >> 5;
  const int hi   = lane >> 4;   // 0 | 1  (lane half)
  const int l15  = lane & 15;
  const int b0   = blockIdx.x * 16;

  // ---- init: alpha0 = startT + em[0];  stage em[1] into sEm[0] ----
  {
    int base = tid * 16;
    int r = base >> 7, c = base & 127;
    const float* e0 = em + (0 * B + b0 + r) * T + c;
#pragma unroll
    for (int j = 0; j < 16; ++j) sAlpha[r][c + j] = startT[c + j] + e0[j];

    const float4* src = (const float4*)(em + (1 * B + b0) * T);
    float4* dst = (float4*)&sEm[0][0][0];
#pragma unroll
    for (int j = 0; j < 4; ++j) dst[tid + j * 128] = src[tid + j * 128];
  }

  // ---- hoist B-fragments of exp(transitions) into registers ----
  // Wave covers N columns [n0, n0+32): two 16-wide tiles.
  // B-frag layout (4x16 f32): lane half hi, VGPR v -> K = 4*kk + v + 2*hi,
  //                           lane&15 -> N. expTt is N-major so K is contiguous.
  const int    n0  = wave * 32;
  const float* bt0 = expTt + (n0 + l15) * T;
  const float* bt1 = expTt + (n0 + 16 + l15) * T;
  v2f bf0[32], bf1[32];
#pragma unroll
  for (int kk = 0; kk < 32; ++kk) {
    int ka = 4 * kk + 2 * hi;
    bf0[kk] = *(const v2f*)&bt0[ka];
    bf1[kk] = *(const v2f*)&bt1[ka];
  }
  __syncthreads();

  int buf = 0;
  for (int s = 1; s < S; ++s) {
    // ---- issue next-step emission loads (register staging) ----
    float4 pre0, pre1, pre2, pre3;
    const bool havePre = (s + 1 < S);
    if (havePre) {
      const float4* src = (const float4*)(em + ((s + 1) * B + b0) * T);
      pre0 = src[tid];       pre1 = src[tid + 128];
      pre2 = src[tid + 256]; pre3 = src[tid + 384];
    }
    if (s + 2 < S) {  // warm L2 two steps out -> global_prefetch_b8
      const char* p = (const char*)(em + ((s + 2) * B + b0) * T);
      __builtin_prefetch(p + tid * 64, 0, 1);
    }

    // ---- phase A: row max of alpha ----
    {
      int r = tid >> 3, ch = tid & 7, cb = ch * 16;
      float m = -3.402823466e+38f;
#pragma unroll
      for (int j = 0; j < 16; ++j) m = fmaxf(m, sAlpha[r][cb + j]);
      sPart[r][ch] = m;
    }
    __syncthreads();
    if (tid < 16) {
      float m = sPart[tid][0];
#pragma unroll
      for (int j = 1; j < 8; ++j) m = fmaxf(m, sPart[tid][j]);
      sAmax[tid] = m;
      sMask[tid] = maskp[s * B + b0 + tid];
    }
    __syncthreads();

    // ---- phase B: expA = exp(alpha - rowmax) ----
    {
      int base = tid * 16;
      int r = base >> 7, c = base & 127;
      float m = sAmax[r];
#pragma unroll
      for (int j = 0; j < 16; ++j)
        sExpA[r][c + j] = __expf(sAlpha[r][c + j] - m);
    }
    __syncthreads();

    // ---- phase C: P = expA @ expT  (fp32 WMMA; B resident in VGPRs) ----
    v8f acc0 = {};
    v8f acc1 = {};
#pragma unroll
    for (int kk = 0; kk < 32; ++kk) {
      v2f a = *(const v2f*)&sExpA[l15][4 * kk + 2 * hi];
      acc0 = __builtin_amdgcn_wmma_f32_16x16x4_f32(
          false, a, false, bf0[kk], (short)0, acc0, false, false);
      acc1 = __builtin_amdgcn_wmma_f32_16x16x4_f32(
          false, a, false, bf1[kk], (short)0, acc1, false, false);
    }

    // ---- phase D: alpha = rowmax + log(P) + em[s]  (masked) ----
#pragma unroll
    for (int k = 0; k < 8; ++k) {
      int M = k + 8 * hi;          // C/D layout: vgpr k, lane half -> row
      int N = n0 + l15;
      float am = sAmax[M];
      float v0 = am + __logf(acc0[k]) + sEm[buf][M][N];
      float v1 = am + __logf(acc1[k]) + sEm[buf][M][N + 16];
      if (sMask[M]) {
        sAlpha[M][N]      = v0;
        sAlpha[M][N + 16] = v1;
      }
    }
    __syncthreads();

    // ---- commit staged emissions into the other buffer ----
    if (havePre) {
      float4* dst = (float4*)&sEm[buf ^ 1][0][0];
      dst[tid]       = pre0; dst[tid + 128] = pre1;
      dst[tid + 256] = pre2; dst[tid + 384] = pre3;
      buf ^= 1;
    }
    __syncthreads();
  }

  // ---- final: norm[b] = logsumexp_j(alpha[b,j] + endT[j]) ----
  {
    int r = tid >> 3, ch = tid & 7, cb = ch * 16;
    float m = -3.402823466e+38f;
#pragma unroll
    for (int j = 0; j < 16; ++j) m = fmaxf(m, sAlpha[r][cb + j] + endT[cb + j]);
    sPart[r][ch] = m;
  }
  __syncthreads();
  if (tid < 16) {
    float m = sPart[tid][0];
#pragma unroll
    for (int j = 1; j < 8; ++j) m = fmaxf(m, sPart[tid][j]);
    sAmax[tid] = m;
  }
  __syncthreads();
  {
    int r = tid >> 3, ch = tid & 7, cb = ch * 16;
    float m = sAmax[r];
    float ssum = 0.f;
#pragma unroll
    for (int j = 0; j < 16; ++j)
      ssum += __expf(sAlpha[r][cb + j] + endT[cb + j] - m);
    sPart[r][ch] = ssum;
  }
  __syncthreads();
  if (tid < 16) {
    float ssum = 0.f;
#pragma unroll
    for (int j = 0; j < 8; ++j) ssum += sPart[tid][j];
    normOut[b0 + tid] = sAmax[tid] + __logf(ssum);
  }
}

// ---------------------------------------------------------------------------
// Kernel 2: numerator score, one thread per batch element.
// ---------------------------------------------------------------------------
__global__ __launch_bounds__(256) void crf_score_kernel(
    const float* __restrict__ em, const long long* __restrict__ tags,
    const unsigned char* __restrict__ maskp, const float* __restrict__ startT,
    const float* __restrict__ endT, const float* __restrict__ trans,
    float* __restrict__ scoreOut) {
  int b = threadIdx.x;
  long long prev = tags[b];
  float sc = startT[prev] + em[(0 * B + b) * T + (int)prev];
  long long last = prev;
  for (int s = 1; s < S; ++s) {
    long long cur = tags[s * B + b];
    unsigned char mk = maskp[s * B + b];
    float m = mk ? 1.0f : 0.0f;
    sc += (trans[(int)prev * T + (int)cur] + em[(s * B + b) * T + (int)cur]) * m;
    if (mk) last = cur;
    prev = cur;
  }
  sc += endT[last];
  scoreOut[b] = sc;
}

// ---------------------------------------------------------------------------
// Kernel 3: llh = sum_b (score[b] - norm[b])
// ---------------------------------------------------------------------------
__global__ __launch_bounds__(256) void crf_reduce_kernel(
    const float* __restrict__ score, const float* __restrict__ norm,
    float* __restrict__ out) {
  __shared__ float red[256];
  int t = threadIdx.x;
  red[t] = score[t] - norm[t];
  __syncthreads();
  for (int off = 128; off > 0; off >>= 1) {
    if (t < off) red[t] += red[t + off];
    __syncthreads();
  }
  if (t == 0) out[0] = red[0];
}

// ---------------------------------------------------------------------------
extern "C" void kernel_launch(void* const* d_in, const int* in_sizes, int n_in,
                              void* d_out, int out_size, void* d_ws, size_t ws_size,
                              hipStream_t stream) {
  (void)in_sizes; (void)n_in; (void)out_size; (void)ws_size;
  const float*         em     = (const float*)d_in[0];
  const long long*     tags   = (const long long*)d_in[1];
  const unsigned char* maskp  = (const unsigned char*)d_in[2];
  const float*         startT = (const float*)d_in[3];
  const float*         endT   = (const float*)d_in[4];
  const float*         trans  = (const float*)d_in[5];

  float* wsf   = (float*)d_ws;
  float* norm  = wsf;          // 256 floats
  float* score = wsf + 256;    // 256 floats
  float* expTt = wsf + 512;    // 128*128 floats

  crf_exp_trans_kernel<<<(T * T) / 256, 256, 0, stream>>>(trans, expTt);
  crf_forward_kernel<<<B / 16, 128, 0, stream>>>(em, maskp, startT, endT,
                                                 expTt, norm);
  crf_score_kernel<<<1, 256, 0, stream>>>(em, tags, maskp, startT, endT,
                                          trans, score);
  crf_reduce_kernel<<<1, 256, 0, stream>>>(score, norm, (float*)d_out);
}